// GraphSAGE_72739566125841
// MI455X (gfx1250) — compile-verified
//
#include <hip/hip_runtime.h>
#include <hip/hip_bf16.h>

typedef __attribute__((ext_vector_type(2))) float v2f;
typedef __attribute__((ext_vector_type(8))) float v8f;

#define IN_D 64
#define ROWS_PB 64        // rows per block (4 row-tiles of 16)
#define XS_STRIDE 68      // 272B row stride: 16B-aligned + 4-bank rotation per row

// ---------------- zero fill (float4 granularity) ----------------
__global__ __launch_bounds__(256) void zero_f4(float4* __restrict__ p, long long n4) {
    long long t = (long long)blockIdx.x * blockDim.x + threadIdx.x;
    if (t < n4) p[t] = make_float4(0.f, 0.f, 0.f, 0.f);
}

// ---------------- degree count ----------------
__global__ __launch_bounds__(256) void deg_count(const int* __restrict__ dst,
                                                 float* __restrict__ deg, int n_edges) {
    int t = blockIdx.x * blockDim.x + threadIdx.x;
    if (t < n_edges) atomicAdd(&deg[dst[t]], 1.0f);
}

__global__ __launch_bounds__(256) void finish_inv(float* __restrict__ inv, int n) {
    int t = blockIdx.x * blockDim.x + threadIdx.x;
    if (t < n) inv[t] = 1.0f / (inv[t] + 1.0f);
}

// ---------------- edge scatter: neigh[dst] += h[src] ----------------
// one thread per (edge, float4-chunk); 16 chunks cover 64 feats
__global__ __launch_bounds__(256) void scatter_add(const int* __restrict__ src,
                                                   const int* __restrict__ dst,
                                                   const float* __restrict__ h,
                                                   float* __restrict__ neigh,
                                                   int n_edges) {
    long long t = (long long)blockIdx.x * blockDim.x + threadIdx.x;
    int e = (int)(t >> 4);
    int c = (int)(t & 15);
    if (e >= n_edges) return;
    int s = src[e];
    int d = dst[e];
    const float4 v = ((const float4*)(h + (size_t)s * IN_D))[c];
    float* o = neigh + (size_t)d * IN_D + c * 4;
    atomicAdd(o + 0, v.x);
    atomicAdd(o + 1, v.y);
    atomicAdd(o + 2, v.z);
    atomicAdd(o + 3, v.w);
}

// ---------------- fused combine + GEMM via V_WMMA_F32_16X16X4_F32 ----------------
// out[i][o] = act( sum_k ((h[i][k]+neigh[i][k])*inv[i]) * W[o][k] + b[o] )
// block = 128 threads (4 waves); block handles 64 rows (4 row-tiles);
// wave w handles out cols [16w,16w+16) across all 4 row-tiles.
// B fragments are hoisted out of the k-loop (branchless zero-masking), so the
// inner loop is 4 ds_load_b64 + 4 independent v_wmma chains per k-step.
__global__ __launch_bounds__(128)
void sage_gemm(const float* __restrict__ h, const float* __restrict__ neigh,
               const float* __restrict__ inv, const float* __restrict__ W,
               const float* __restrict__ bias, float* __restrict__ out,
               int n, int out_d, int do_relu) {
    __shared__ float xs[ROWS_PB * XS_STRIDE];

    const int row0 = blockIdx.x * ROWS_PB;
    const int tid  = threadIdx.x;

    // ---- stage combined 64x64 tile into LDS (2 threads per row) ----
    {
        const int r  = tid >> 1;            // 0..63
        const int cb = (tid & 1) * 32;      // 0 or 32
        const int grow = row0 + r;
        float4* dptr = (float4*)&xs[r * XS_STRIDE + cb];
        if (grow < n) {
            const float s = inv[grow];
            const float4* hp = (const float4*)(h + (size_t)grow * IN_D + cb);
            const float4* np = (const float4*)(neigh + (size_t)grow * IN_D + cb);
#pragma unroll
            for (int j = 0; j < 8; ++j) {
                float4 a = hp[j], b = np[j], x;
                x.x = (a.x + b.x) * s;  x.y = (a.y + b.y) * s;
                x.z = (a.z + b.z) * s;  x.w = (a.w + b.w) * s;
                dptr[j] = x;
            }
        } else {
#pragma unroll
            for (int j = 0; j < 8; ++j) dptr[j] = make_float4(0.f, 0.f, 0.f, 0.f);
        }
    }
    __syncthreads();

    const int wave = tid >> 5;            // 0..3 -> column tile
    const int lane = tid & 31;
    const int col0 = wave * 16;
    if (col0 >= ((out_d + 15) & ~15)) return;   // wave-uniform exit (EXEC stays full)

    const int m16  = lane & 15;
    const int half = lane >> 4;

    // ---- hoist B fragments (W tile), branchless zero for padded rows ----
    const int  wrow   = col0 + m16;
    const bool wvalid = wrow < out_d;
    const float* wp   = W + (size_t)(wvalid ? wrow : 0) * IN_D;   // clamped, always valid
    const float wm    = wvalid ? 1.f : 0.f;
    v2f bfrag[16];
#pragma unroll
    for (int kt = 0; kt < 16; ++kt) {
        v2f braw = *(const v2f*)(wp + kt * 4 + half * 2);
        bfrag[kt].x = braw.x * wm;
        bfrag[kt].y = braw.y * wm;
    }

    // ---- 4 independent accumulator chains over K=64 ----
    v8f acc[4] = {v8f{}, v8f{}, v8f{}, v8f{}};
#pragma unroll
    for (int kt = 0; kt < 16; ++kt) {
        const int ko = kt * 4 + half * 2;   // A lane half holds K = 2h, 2h+1
#pragma unroll
        for (int rt = 0; rt < 4; ++rt) {
            v2f a = *(const v2f*)&xs[(rt * 16 + m16) * XS_STRIDE + ko];
            acc[rt] = __builtin_amdgcn_wmma_f32_16x16x4_f32(
                /*neg_a=*/false, a, /*neg_b=*/false, bfrag[kt],
                /*c_mod=*/(short)0, acc[rt], /*reuse_a=*/false, /*reuse_b=*/false);
        }
    }

    // ---- epilogue: bias + relu + store (D: vgpr r -> M = 8*half + r, N = lane%16) ----
    const int  ocol   = col0 + m16;
    const bool cvalid = ocol < out_d;
    const float bv    = cvalid ? bias[ocol] : 0.f;
#pragma unroll
    for (int rt = 0; rt < 4; ++rt) {
#pragma unroll
        for (int r = 0; r < 8; ++r) {
            const int grow = row0 + rt * 16 + half * 8 + r;
            if (cvalid && grow < n) {
                float v = acc[rt][r] + bv;
                if (do_relu) v = v > 0.f ? v : 0.f;
                out[(size_t)grow * out_d + ocol] = v;
            }
        }
    }
}

// ---------------- host orchestration ----------------
extern "C" void kernel_launch(void* const* d_in, const int* in_sizes, int n_in,
                              void* d_out, int out_size, void* d_ws, size_t ws_size,
                              hipStream_t stream) {
    const float* feats = (const float*)d_in[0];
    const int*   src   = (const int*)d_in[1];
    const int*   dst   = (const int*)d_in[2];
    const float* W0 = (const float*)d_in[3];
    const float* b0 = (const float*)d_in[4];
    const float* W1 = (const float*)d_in[5];
    const float* b1 = (const float*)d_in[6];
    const float* W2 = (const float*)d_in[7];
    const float* b2 = (const float*)d_in[8];

    const int N = in_sizes[0] / IN_D;   // 100000
    const int E = in_sizes[1];          // 1600000
    const int N_CLS = in_sizes[8];      // 40 (size of b2)

    float* ws    = (float*)d_ws;
    float* invb  = ws;                          // N floats (deg -> inv, in place)
    float* neigh = invb + (size_t)N;            // N*64
    float* h1    = neigh + (size_t)N * IN_D;    // N*64
    float* h2    = h1 + (size_t)N * IN_D;       // N*64

    const int THR = 256;
    // degree + inverse
    {
        long long n4 = (long long)N / 4;
        zero_f4<<<(int)((n4 + THR - 1) / THR), THR, 0, stream>>>((float4*)invb, n4);
        deg_count<<<(E + THR - 1) / THR, THR, 0, stream>>>(dst, invb, E);
        finish_inv<<<(N + THR - 1) / THR, THR, 0, stream>>>(invb, N);
    }

    const long long feat4 = (long long)N * IN_D / 4;
    const int zeroBlocks = (int)((feat4 + THR - 1) / THR);
    const long long scatT = (long long)E * 16;
    const int scatBlocks = (int)((scatT + THR - 1) / THR);
    const int gemmBlocks = (N + ROWS_PB - 1) / ROWS_PB;

    // layer 0: features -> h1
    zero_f4<<<zeroBlocks, THR, 0, stream>>>((float4*)neigh, feat4);
    scatter_add<<<scatBlocks, THR, 0, stream>>>(src, dst, feats, neigh, E);
    sage_gemm<<<gemmBlocks, 128, 0, stream>>>(feats, neigh, invb, W0, b0, h1, N, IN_D, 1);

    // layer 1: h1 -> h2
    zero_f4<<<zeroBlocks, THR, 0, stream>>>((float4*)neigh, feat4);
    scatter_add<<<scatBlocks, THR, 0, stream>>>(src, dst, h1, neigh, E);
    sage_gemm<<<gemmBlocks, 128, 0, stream>>>(h1, neigh, invb, W1, b1, h2, N, IN_D, 1);

    // layer 2: h2 -> d_out (no relu)
    zero_f4<<<zeroBlocks, THR, 0, stream>>>((float4*)neigh, feat4);
    scatter_add<<<scatBlocks, THR, 0, stream>>>(src, dst, h2, neigh, E);
    sage_gemm<<<gemmBlocks, 128, 0, stream>>>(h2, neigh, invb, W2, b2, (float*)d_out,
                                              N, N_CLS, 0);
}